// GCN_65068754534584
// MI455X (gfx1250) — compile-verified
//
#include <hip/hip_runtime.h>
#include <hip/hip_bf16.h>

typedef __attribute__((ext_vector_type(2))) float v2f;
typedef __attribute__((ext_vector_type(8))) float v8f;

#define N_NODES 100000
#define F_IN    512
#define H_DIM   16
#define C_DIM   40
#define N_EDGES 3200000
#define EPS_BN  1e-5f
#define N_TILES (N_NODES / 16)   // 6250, exact

// ---------------------------------------------------------------- degree / dis
__global__ void k_init_deg(float* __restrict__ deg) {
    int i = blockIdx.x * blockDim.x + threadIdx.x;
    if (i < N_NODES) deg[i] = 1.0f;            // self loop
}

__global__ void k_edge_deg(const int* __restrict__ col, float* __restrict__ deg) {
    int e = blockIdx.x * blockDim.x + threadIdx.x;
    if (e < N_EDGES) atomicAdd(&deg[col[e]], 1.0f);
}

__global__ void k_deg_to_dis(float* __restrict__ deg) {
    int i = blockIdx.x * blockDim.x + threadIdx.x;
    if (i < N_NODES) deg[i] = rsqrtf(deg[i]);  // deg >= 1 always
}

// ---------------------------------------------------------------- GEMM1: x @ W1  (N x 512) * (512 x 16)
__global__ void k_gemm1(const float* __restrict__ x, const float* __restrict__ W1,
                        float* __restrict__ xw1) {
    int wave = (blockIdx.x * blockDim.x + threadIdx.x) >> 5;
    int lane = threadIdx.x & 31;
    if (wave >= N_TILES) return;               // wave-uniform: EXEC stays all-1s
    const int m0   = wave * 16;
    const int half = lane >> 4;                // 0: K pair {0,1}, 1: K pair {2,3}
    const int lid  = lane & 15;

    v8f c = {};
    const float* arow = x  + (size_t)(m0 + lid) * F_IN + 2 * half;  // 8B aligned
    const float* bcol = W1 + (size_t)(2 * half) * H_DIM + lid;

    for (int kb = 0; kb < F_IN; kb += 4) {
        v2f a = *(const v2f*)(arow + kb);                    // A[m][kb+2h .. +1]
        v2f b;
        b.x = bcol[kb * H_DIM];                              // B[kb+2h  ][n]
        b.y = bcol[kb * H_DIM + H_DIM];                      // B[kb+2h+1][n]
        c = __builtin_amdgcn_wmma_f32_16x16x4_f32(false, a, false, b,
                                                  (short)0, c, false, false);
    }
    float* orow = xw1 + (size_t)m0 * H_DIM;
    #pragma unroll
    for (int r = 0; r < 8; ++r)
        orow[(size_t)(r + 8 * half) * H_DIM + lid] = c[r];
}

// ---------------------------------------------------------------- self-loop seed (layer 1)
__global__ void k_self1(const float* __restrict__ dis, const float* __restrict__ xw1,
                        float* __restrict__ h) {
    int t = blockIdx.x * blockDim.x + threadIdx.x;
    if (t >= N_NODES * H_DIM) return;
    int i = t >> 4;
    float d = dis[i];
    h[t] = d * d * xw1[t];
}

// ---------------------------------------------------------------- edge scatter (templated width)
template<int W>
__global__ void k_edge_agg(const int* __restrict__ row, const int* __restrict__ col,
                           const float* __restrict__ dis, const float* __restrict__ src,
                           float* __restrict__ dst) {
    long long t = (long long)blockIdx.x * blockDim.x + threadIdx.x;
    if (t >= (long long)N_EDGES * W) return;
    int e = (int)(t / W);
    int j = (int)(t % W);
    int r = row[e], c = col[e];
    float w = dis[r] * dis[c];
    atomicAdd(&dst[(size_t)c * W + j], w * src[(size_t)r * W + j]);
}

// ---------------------------------------------------------------- bias + relu (in place)
__global__ void k_bias_relu(float* __restrict__ h, const float* __restrict__ b1) {
    int t = blockIdx.x * blockDim.x + threadIdx.x;
    if (t >= N_NODES * H_DIM) return;
    h[t] = fmaxf(h[t] + b1[t & 15], 0.0f);
}

// ---------------------------------------------------------------- batch-norm stats
__global__ void k_zero_stats(float* __restrict__ stats) {
    if (threadIdx.x < 32) stats[threadIdx.x] = 0.0f;
}

__global__ void k_stats(const float* __restrict__ h, float* __restrict__ stats) {
    __shared__ float s_sum[256], s_sq[256];
    int j = threadIdx.x & 15;
    int g = threadIdx.x >> 4;                  // 16 row-groups per block
    float s = 0.0f, q = 0.0f;
    for (int i = blockIdx.x * 16 + g; i < N_NODES; i += gridDim.x * 16) {
        float v = h[(size_t)i * H_DIM + j];
        s += v; q += v * v;
    }
    s_sum[threadIdx.x] = s; s_sq[threadIdx.x] = q;
    __syncthreads();
    if (g == 0) {
        float ts = 0.0f, tq = 0.0f;
        #pragma unroll
        for (int k = 0; k < 16; ++k) { ts += s_sum[k * 16 + j]; tq += s_sq[k * 16 + j]; }
        atomicAdd(&stats[j], ts);
        atomicAdd(&stats[16 + j], tq);
    }
}

__global__ void k_bn_apply(float* __restrict__ h, const float* __restrict__ stats,
                           const float* __restrict__ gamma, const float* __restrict__ beta) {
    int t = blockIdx.x * blockDim.x + threadIdx.x;
    if (t >= N_NODES * H_DIM) return;
    int j = t & 15;
    float mean = stats[j] * (1.0f / N_NODES);
    float var  = stats[16 + j] * (1.0f / N_NODES) - mean * mean;
    h[t] = (h[t] - mean) * rsqrtf(var + EPS_BN) * gamma[j] + beta[j];
}

// ---------------------------------------------------------------- GEMM2: h @ W2  (N x 16) * (16 x 40)
__global__ void k_gemm2(const float* __restrict__ h, const float* __restrict__ W2,
                        float* __restrict__ xw2) {
    int wave = (blockIdx.x * blockDim.x + threadIdx.x) >> 5;
    int lane = threadIdx.x & 31;
    if (wave >= N_TILES) return;
    const int m0   = wave * 16;
    const int half = lane >> 4;
    const int lid  = lane & 15;

    v8f c0 = {}, c1 = {}, c2 = {};
    const float* arow = h + (size_t)(m0 + lid) * H_DIM + 2 * half;

    #pragma unroll
    for (int kb = 0; kb < H_DIM; kb += 4) {
        v2f a = *(const v2f*)(arow + kb);
        int k0 = kb + 2 * half;
        int n0 = lid, n1 = 16 + lid, n2 = 32 + lid;
        v2f b0, b1v, b2v;
        b0.x  = W2[k0 * C_DIM + n0];          b0.y  = W2[(k0 + 1) * C_DIM + n0];
        b1v.x = W2[k0 * C_DIM + n1];          b1v.y = W2[(k0 + 1) * C_DIM + n1];
        b2v.x = (n2 < C_DIM) ? W2[k0 * C_DIM + n2] : 0.0f;
        b2v.y = (n2 < C_DIM) ? W2[(k0 + 1) * C_DIM + n2] : 0.0f;
        c0 = __builtin_amdgcn_wmma_f32_16x16x4_f32(false, a, false, b0,  (short)0, c0, false, false);
        c1 = __builtin_amdgcn_wmma_f32_16x16x4_f32(false, a, false, b1v, (short)0, c1, false, false);
        c2 = __builtin_amdgcn_wmma_f32_16x16x4_f32(false, a, false, b2v, (short)0, c2, false, false);
    }
    #pragma unroll
    for (int r = 0; r < 8; ++r) {
        int m = m0 + r + 8 * half;
        xw2[(size_t)m * C_DIM + lid]      = c0[r];
        xw2[(size_t)m * C_DIM + 16 + lid] = c1[r];
        if (32 + lid < C_DIM) xw2[(size_t)m * C_DIM + 32 + lid] = c2[r];
    }
}

// ---------------------------------------------------------------- self-loop seed + bias (layer 2)
__global__ void k_self2(const float* __restrict__ dis, const float* __restrict__ xw2,
                        const float* __restrict__ b2, float* __restrict__ out) {
    long long t = (long long)blockIdx.x * blockDim.x + threadIdx.x;
    if (t >= (long long)N_NODES * C_DIM) return;
    int i = (int)(t / C_DIM);
    int c = (int)(t % C_DIM);
    float d = dis[i];
    out[t] = d * d * xw2[t] + b2[c];
}

// ---------------------------------------------------------------- log_softmax (in place)
__global__ void k_logsoftmax(float* __restrict__ out) {
    int i = blockIdx.x * blockDim.x + threadIdx.x;
    if (i >= N_NODES) return;
    float* row = out + (size_t)i * C_DIM;
    float m = -3.4e38f;
    #pragma unroll
    for (int c = 0; c < C_DIM; ++c) m = fmaxf(m, row[c]);
    float s = 0.0f;
    #pragma unroll
    for (int c = 0; c < C_DIM; ++c) s += __expf(row[c] - m);
    float lse = m + __logf(s);
    #pragma unroll
    for (int c = 0; c < C_DIM; ++c) row[c] -= lse;
}

// ---------------------------------------------------------------- launch
extern "C" void kernel_launch(void* const* d_in, const int* in_sizes, int n_in,
                              void* d_out, int out_size, void* d_ws, size_t ws_size,
                              hipStream_t stream) {
    const float* x     = (const float*)d_in[0];
    const int*   edges = (const int*)  d_in[1];   // [2, E] flattened
    const float* W1    = (const float*)d_in[2];
    const float* b1    = (const float*)d_in[3];
    const float* W2    = (const float*)d_in[4];
    const float* b2    = (const float*)d_in[5];
    const float* gamma = (const float*)d_in[6];
    const float* beta  = (const float*)d_in[7];
    float*       out   = (float*)d_out;

    const int* row = edges;
    const int* col = edges + N_EDGES;

    // workspace layout (floats)
    float* ws    = (float*)d_ws;
    float* dis   = ws;                                  // N       (deg -> dis in place)
    float* xw1   = ws + (size_t)N_NODES;                // N*16
    float* h     = xw1 + (size_t)N_NODES * H_DIM;       // N*16
    float* xw2   = h   + (size_t)N_NODES * H_DIM;       // N*40
    float* stats = xw2 + (size_t)N_NODES * C_DIM;       // 32

    const int B = 256;
    const int gN   = (N_NODES + B - 1) / B;
    const int gE   = (N_EDGES + B - 1) / B;
    const int gNH  = (N_NODES * H_DIM + B - 1) / B;
    const int gNC  = (int)(((long long)N_NODES * C_DIM + B - 1) / B);
    const int gEH  = (int)(((long long)N_EDGES * H_DIM + B - 1) / B);
    const int gEC  = (int)(((long long)N_EDGES * C_DIM + B - 1) / B);
    const int gMM  = (N_TILES + 7) / 8;                 // 8 waves / 256-thread block

    // degrees -> dis
    k_init_deg <<<gN, B, 0, stream>>>(dis);
    k_edge_deg <<<gE, B, 0, stream>>>(col, dis);
    k_deg_to_dis<<<gN, B, 0, stream>>>(dis);

    // layer 1
    k_gemm1    <<<gMM, B, 0, stream>>>(x, W1, xw1);
    k_self1    <<<gNH, B, 0, stream>>>(dis, xw1, h);
    k_edge_agg<H_DIM><<<gEH, B, 0, stream>>>(row, col, dis, xw1, h);
    k_bias_relu<<<gNH, B, 0, stream>>>(h, b1);

    // batch norm
    k_zero_stats<<<1, 32, 0, stream>>>(stats);
    k_stats    <<<512, B, 0, stream>>>(h, stats);
    k_bn_apply <<<gNH, B, 0, stream>>>(h, stats, gamma, beta);

    // layer 2
    k_gemm2    <<<gMM, B, 0, stream>>>(h, W2, xw2);
    k_self2    <<<gNC, B, 0, stream>>>(dis, xw2, b2, out);
    k_edge_agg<C_DIM><<<gEC, B, 0, stream>>>(row, col, dis, xw2, out);
    k_logsoftmax<<<gN, B, 0, stream>>>(out);
}